// ConceptGCN_68693706932806
// MI455X (gfx1250) — compile-verified
//
#include <hip/hip_runtime.h>
#include <cstdint>

typedef __attribute__((ext_vector_type(2))) float v2f;
typedef __attribute__((ext_vector_type(8))) float v8f;

#define FEAT_OUT 256   // both layers emit 256 features

// ---------------------------------------------------------------------------
// Forced no-return HW float atomic (global_atomic_add_f32, STOREcnt path).
// ---------------------------------------------------------------------------
__device__ __forceinline__ void atomic_add_f32(float* p, float v) {
    asm volatile("global_atomic_add_f32 %0, %1, off" :: "v"(p), "v"(v) : "memory");
}

// ---------------------------------------------------------------------------
// Degree / normalization:  deg[n] = 1 (self loop) + #incoming edges
// ---------------------------------------------------------------------------
__global__ __launch_bounds__(256) void k_init_deg(float* __restrict__ deg, int n) {
    int i = blockIdx.x * 256 + threadIdx.x;
    if (i < n) deg[i] = 1.0f;                      // self-loop contribution
}

__global__ __launch_bounds__(256) void k_count_deg(const long long* __restrict__ dst,
                                                   float* __restrict__ deg, int E) {
    int e = blockIdx.x * 256 + threadIdx.x;
    if (e < E) atomic_add_f32(&deg[(int)dst[e]], 1.0f);
}

__global__ __launch_bounds__(256) void k_dinv(float* __restrict__ deg, int n) {
    int i = blockIdx.x * 256 + threadIdx.x;
    if (i < n) {
        float d = deg[i];
        deg[i] = (d > 0.0f) ? rsqrtf(d) : 0.0f;    // in-place deg -> dinv
    }
}

// ---------------------------------------------------------------------------
// Repack W (Kx256 row-major) into K-pair-interleaved layout so a B fragment
// (W[kk][col], W[kk+1][col]) is one aligned 8-byte load:
//   Wp[(k>>1)*512 + col*2 + (k&1)] = W[k*256 + col]
// ---------------------------------------------------------------------------
__global__ __launch_bounds__(256) void k_packW(const float* __restrict__ W,
                                               float* __restrict__ Wp, int total) {
    int i = blockIdx.x * 256 + threadIdx.x;
    if (i >= total) return;
    int k = i >> 8, col = i & 255;
    Wp[((k >> 1) << 9) + (col << 1) + (k & 1)] = W[i];
}

// ---------------------------------------------------------------------------
// WMMA GEMM:  OUT[m][0:256] = (X[m][:] @ W) * dinv[m]
// X: MxK row-major, Wp: packed (see k_packW).  M must be a multiple of 16.
// Block = 128 threads = 4 waves; wave w covers cols [64w, 64w+64); 16 rows/block.
// V_WMMA_F32_16X16X4_F32 fragment layouts (ISA 7.12.2, 32-bit A/B/C):
//   A (16x4): VGPR0/1 = K{0,1} for lanes 0-15, K{2,3} for lanes 16-31; M = lane&15
//   B (4x16): VGPR0/1 = rows K{0,1} (lanes 0-15) / K{2,3} (lanes 16-31); N = lane&15
//   C (16x16): VGPR v -> row v (lanes 0-15) / row v+8 (lanes 16-31); N = lane&15
// ---------------------------------------------------------------------------
template <int K>
__global__ __launch_bounds__(128) void k_gemm_wmma(const float* __restrict__ X,
                                                   const float* __restrict__ Wp,
                                                   const float* __restrict__ dinv,
                                                   float* __restrict__ OUT, int M) {
    const int lane = threadIdx.x & 31;
    const int wave = threadIdx.x >> 5;            // 0..3
    const int row0 = blockIdx.x * 16;
    if (row0 >= M) return;
    const int col0 = wave * 64;
    const int half = lane >> 4;                   // 0: lanes 0-15, 1: lanes 16-31
    const int l16  = lane & 15;

    v8f acc0 = {}, acc1 = {}, acc2 = {}, acc3 = {};

    const float* arow = X + (size_t)(row0 + l16) * K;
    // packed-W base for this lane: k-pair row ((k>>1)+half), col (col0+l16)
    const float* bbase = Wp + (size_t)half * 512 + (size_t)(col0 + l16) * 2;

    for (int k = 0; k < K; k += 4) {
        // A fragment: one 8-byte load per lane
        const float* ap = arow + k + 2 * half;
        v2f a; a.x = ap[0]; a.y = ap[1];

        // B fragments: one aligned 8-byte load per 16x16 tile
        const float* bp = bbase + (size_t)(k >> 1) * 512;
        v2f b0 = *(const v2f*)(bp);
        v2f b1 = *(const v2f*)(bp + 32);
        v2f b2 = *(const v2f*)(bp + 64);
        v2f b3 = *(const v2f*)(bp + 96);

        acc0 = __builtin_amdgcn_wmma_f32_16x16x4_f32(false, a, false, b0, (short)0, acc0, false, false);
        acc1 = __builtin_amdgcn_wmma_f32_16x16x4_f32(false, a, false, b1, (short)0, acc1, false, false);
        acc2 = __builtin_amdgcn_wmma_f32_16x16x4_f32(false, a, false, b2, (short)0, acc2, false, false);
        acc3 = __builtin_amdgcn_wmma_f32_16x16x4_f32(false, a, false, b3, (short)0, acc3, false, false);
    }

    // Epilogue: scale row m by dinv[m] and store (fused src-side normalization)
#pragma unroll
    for (int v = 0; v < 8; ++v) {
        const int r = row0 + v + 8 * half;
        const float s = dinv[r];
        float* orow = OUT + (size_t)r * FEAT_OUT + col0 + l16;
        orow[0]  = acc0[v] * s;
        orow[16] = acc1[v] * s;
        orow[32] = acc2[v] * s;
        orow[48] = acc3[v] * s;
    }
}

// ---------------------------------------------------------------------------
// Seed copy (acc[n] = hs[n], the self-loop message), float4-wide
// ---------------------------------------------------------------------------
__global__ __launch_bounds__(256) void k_copy4(float4* __restrict__ dst,
                                               const float4* __restrict__ src, int n4) {
    int i = blockIdx.x * 256 + threadIdx.x;
    if (i < n4) dst[i] = src[i];
}

// ---------------------------------------------------------------------------
// Edge scatter: acc[dst][c] += hs[src][c].  One wave per edge; lane handles
// 8 columns -> fully coalesced 1KB row gather + 1KB of no-return f32 atomics.
// ---------------------------------------------------------------------------
__global__ __launch_bounds__(256) void k_scatter(const long long* __restrict__ src,
                                                 const long long* __restrict__ dst,
                                                 const float* __restrict__ hs,
                                                 float* __restrict__ acc, int E) {
    int e = blockIdx.x * 8 + (threadIdx.x >> 5);
    if (e >= E) return;
    const int lane = threadIdx.x & 31;
    const float* hp = hs + (size_t)(int)src[e] * FEAT_OUT;
    float* ap = acc + (size_t)(int)dst[e] * FEAT_OUT;
#pragma unroll
    for (int j = 0; j < 8; ++j) {
        const int c = lane + 32 * j;
        atomic_add_f32(ap + c, hp[c]);
    }
}

// ---------------------------------------------------------------------------
// Layer-1 finalize: h[n][c] = relu(dinv[n] * acc[n][c] + b[c])
// ---------------------------------------------------------------------------
__global__ __launch_bounds__(256) void k_relu_bias(const float4* __restrict__ acc,
                                                   const float* __restrict__ dinv,
                                                   const float4* __restrict__ bias,
                                                   float4* __restrict__ h, int n4) {
    int i = blockIdx.x * 256 + threadIdx.x;
    if (i >= n4) return;
    const int row = i >> 6;                 // 64 float4 per 256-wide row
    const int c4  = i & 63;
    const float s = dinv[row];
    float4 a = acc[i], b = bias[c4], r;
    r.x = fmaxf(fmaf(s, a.x, b.x), 0.0f);
    r.y = fmaxf(fmaf(s, a.y, b.y), 0.0f);
    r.z = fmaxf(fmaf(s, a.z, b.z), 0.0f);
    r.w = fmaxf(fmaf(s, a.w, b.w), 0.0f);
    h[i] = r;
}

// ---------------------------------------------------------------------------
// Layer-2 finalize: out[n][c] = dinv[n] * acc[n][c] + b[c]   (no relu)
// ---------------------------------------------------------------------------
__global__ __launch_bounds__(256) void k_final_bias(const float4* __restrict__ acc,
                                                    const float* __restrict__ dinv,
                                                    const float4* __restrict__ bias,
                                                    float4* __restrict__ out, int n4) {
    int i = blockIdx.x * 256 + threadIdx.x;
    if (i >= n4) return;
    const int row = i >> 6;
    const int c4  = i & 63;
    const float s = dinv[row];
    float4 a = acc[i], b = bias[c4], r;
    r.x = fmaf(s, a.x, b.x);
    r.y = fmaf(s, a.y, b.y);
    r.z = fmaf(s, a.z, b.z);
    r.w = fmaf(s, a.w, b.w);
    out[i] = r;
}

// ---------------------------------------------------------------------------
extern "C" void kernel_launch(void* const* d_in, const int* in_sizes, int n_in,
                              void* d_out, int out_size, void* d_ws, size_t ws_size,
                              hipStream_t stream) {
    const float*     x   = (const float*)d_in[0];
    const long long* ei  = (const long long*)d_in[1];   // int64 edge_index [2,E]
    const float*     W1  = (const float*)d_in[2];
    const float*     b1  = (const float*)d_in[3];
    const float*     W2  = (const float*)d_in[4];
    const float*     b2  = (const float*)d_in[5];
    float*           out = (float*)d_out;

    const int N = in_sizes[0] / 768;     // 50000
    const int E = in_sizes[1] / 2;       // 800000
    const long long* esrc = ei;
    const long long* edst = ei + E;

    // workspace layout: bufA = N*256 f32 (51.2MB) | dinv = N f32 | Wp = 768*256 f32
    float* bufA = (float*)d_ws;
    float* dinv = bufA + (size_t)N * FEAT_OUT;
    float* Wp   = dinv + N;

    const int n4 = N * (FEAT_OUT / 4);            // float4 count of a feature map
    const int gN   = (N + 255) / 256;
    const int gE   = (E + 255) / 256;
    const int gEw  = (E + 7) / 8;                 // one wave per edge
    const int gMap = (n4 + 255) / 256;
    const int gGemm = N / 16;                     // 50000/16 = 3125 exact

    // --- symmetric normalization coefficients -----------------------------
    k_init_deg <<<gN, 256, 0, stream>>>(dinv, N);
    k_count_deg<<<gE, 256, 0, stream>>>(edst, dinv, E);
    k_dinv     <<<gN, 256, 0, stream>>>(dinv, N);

    // --- layer 1: hs1 = (x @ W1) * dinv[row]  -> bufA ---------------------
    k_packW<<<(768 * 256 + 255) / 256, 256, 0, stream>>>(W1, Wp, 768 * 256);
    k_gemm_wmma<768><<<gGemm, 128, 0, stream>>>(x, Wp, dinv, bufA, N);
    // acc1 (in d_out) = self-loop seed + edge scatter
    k_copy4  <<<gMap, 256, 0, stream>>>((float4*)out, (const float4*)bufA, n4);
    k_scatter<<<gEw, 256, 0, stream>>>(esrc, edst, bufA, out, E);
    // h2in (bufA) = relu(dinv * acc1 + b1)
    k_relu_bias<<<gMap, 256, 0, stream>>>((const float4*)out, dinv, (const float4*)b1,
                                          (float4*)bufA, n4);

    // --- layer 2: gs2 = (h2in @ W2) * dinv[row]  -> d_out -----------------
    k_packW<<<(256 * 256 + 255) / 256, 256, 0, stream>>>(W2, Wp, 256 * 256);
    k_gemm_wmma<256><<<gGemm, 128, 0, stream>>>(bufA, Wp, dinv, out, N);
    // acc2 (in bufA) = self-loop seed + edge scatter
    k_copy4  <<<gMap, 256, 0, stream>>>((float4*)bufA, (const float4*)out, n4);
    k_scatter<<<gEw, 256, 0, stream>>>(esrc, edst, out, bufA, E);
    // out = dinv * acc2 + b2
    k_final_bias<<<gMap, 256, 0, stream>>>((const float4*)bufA, dinv, (const float4*)b2,
                                           (float4*)out, n4);
}